// _ProposalTargetLayer_63007170232552
// MI455X (gfx1250) — compile-verified
//
#include <hip/hip_runtime.h>
#include <stdint.h>

#define B_ 16
#define N_ 8192
#define K_ 128
#define M_ (N_ + K_)
#define FG_THRESH 0.5f

// ---------------------------------------------------------------------------
// Stage the K_*5 floats of gt_boxes[b] into LDS using the CDNA5 async
// global->LDS DMA path (gfx1250: global_load_async_to_lds_b128, ASYNCcnt).
// 2560 bytes = 160 x 16B transfers; lanes tid<160 each move one 16B chunk.
// ---------------------------------------------------------------------------
__device__ __forceinline__ void stage_gt_async(const float* __restrict__ gt_b,
                                               float* sgt, int tid) {
  if (tid < (K_ * 5 * 4) / 16) {                       // 160 lanes
    // Low 32 bits of a flat pointer to LDS are the wave-relative LDS address.
    uint32_t lds_addr = (uint32_t)(uintptr_t)(sgt) + (uint32_t)tid * 16u;
    uint32_t voff     = (uint32_t)tid * 16u;
    asm volatile("global_load_async_to_lds_b128 %0, %1, %2"
                 :
                 : "v"(lds_addr), "v"(voff), "s"(gt_b)
                 : "memory");
  }
  asm volatile("s_wait_asynccnt 0x0" ::: "memory");    // per-wave DMA complete
  __syncthreads();                                     // visible to all waves
}

// ---------------------------------------------------------------------------
// Kernel 1: per-roi max IoU + first-argmax over K gt boxes; also emits the
// original-order focal inside/outside weights. 2 rois per thread so each
// LDS broadcast of a gt box (1x ds_load_b128 + 1x ds_load_b64) feeds two
// IoU evaluations and two independent divide chains.
// ---------------------------------------------------------------------------
#define ROIS_PER_BLOCK 512   // 256 threads x 2 rois

__global__ void __launch_bounds__(256)
iou_argmax_kernel(const float* __restrict__ all_rois,   // (B,N,5)
                  const float* __restrict__ gt_boxes,   // (B,K,5)
                  float* __restrict__ max_ov,           // (B,M) ws
                  int*   __restrict__ gt_assign,        // (B,M) ws
                  float* __restrict__ inside_w,         // (B,M,4) out
                  float* __restrict__ outside_w)        // (B,M,4) out
{
  __shared__ __align__(16) float  sgt[K_ * 5];  // packed async-DMA landing pad
  __shared__ __align__(16) float4 sg4[K_];      // {x1,y1,x2,y2} per gt
  __shared__ __align__(8)  float2 gad[K_];      // {area, zero-flag} per gt

  const int b   = blockIdx.y;
  const int tid = threadIdx.x;

  stage_gt_async(gt_boxes + (size_t)b * K_ * 5, sgt, tid);

  if (tid < K_) {
    const float x1 = sgt[tid * 5 + 0], y1 = sgt[tid * 5 + 1];
    const float x2 = sgt[tid * 5 + 2], y2 = sgt[tid * 5 + 3];
    const float gw = x2 - x1 + 1.0f,   gh = y2 - y1 + 1.0f;
    sg4[tid] = make_float4(x1, y1, x2, y2);
    gad[tid] = make_float2(gw * gh,
                           ((gw == 1.0f) && (gh == 1.0f)) ? 1.0f : 0.0f);
  }
  __syncthreads();

  const int i0 = blockIdx.x * ROIS_PER_BLOCK + tid;
  const int i1 = i0 + 256;
  const bool v0 = (i0 < M_);
  const bool v1 = (i1 < M_);

  // Fetch both roi boxes (appended-gt rois come from LDS; OOB -> dummy gt0)
  float4 r0, r1;
  {
    if (!v0)           r0 = sg4[0];
    else if (i0 < N_) { const float* p = all_rois + ((size_t)b * N_ + i0) * 5;
                        r0 = make_float4(p[1], p[2], p[3], p[4]); }
    else               r0 = sg4[i0 - N_];
    if (!v1)           r1 = sg4[0];
    else if (i1 < N_) { const float* p = all_rois + ((size_t)b * N_ + i1) * 5;
                        r1 = make_float4(p[1], p[2], p[3], p[4]); }
    else               r1 = sg4[i1 - N_];
  }

  const float aw0 = r0.z - r0.x + 1.0f, ah0 = r0.w - r0.y + 1.0f;
  const float aw1 = r1.z - r1.x + 1.0f, ah1 = r1.w - r1.y + 1.0f;
  const float area0 = aw0 * ah0, area1 = aw1 * ah1;
  const bool az0 = (aw0 == 1.0f) && (ah0 == 1.0f);   // loop-invariant,
  const bool az1 = (aw1 == 1.0f) && (ah1 == 1.0f);   // applied post-loop

  float best0 = -3.0e38f, best1 = -3.0e38f;
  int   bk0 = 0, bk1 = 0;
#pragma unroll 4
  for (int k = 0; k < K_; ++k) {
    const float4 g  = sg4[k];   // ds_load_b128 broadcast
    const float2 az = gad[k];   // ds_load_b64 broadcast
    // roi 0
    {
      const float iw = fmaxf(fminf(r0.z, g.z) - fmaxf(r0.x, g.x) + 1.0f, 0.0f);
      const float ih = fmaxf(fminf(r0.w, g.w) - fmaxf(r0.y, g.y) + 1.0f, 0.0f);
      const float inter = iw * ih;
      float ov = inter / (area0 + az.x - inter);
      ov = (az.y != 0.0f) ? 0.0f : ov;            // zero-area gt
      if (ov > best0) { best0 = ov; bk0 = k; }    // first-max (jnp.argmax)
    }
    // roi 1
    {
      const float iw = fmaxf(fminf(r1.z, g.z) - fmaxf(r1.x, g.x) + 1.0f, 0.0f);
      const float ih = fmaxf(fminf(r1.w, g.w) - fmaxf(r1.y, g.y) + 1.0f, 0.0f);
      const float inter = iw * ih;
      float ov = inter / (area1 + az.x - inter);
      ov = (az.y != 0.0f) ? 0.0f : ov;
      if (ov > best1) { best1 = ov; bk1 = k; }
    }
  }
  // zero-area anchor: whole ov row is -1 -> max=-1, argmax=0
  if (az0) { best0 = -1.0f; bk0 = 0; }
  if (az1) { best1 = -1.0f; bk1 = 0; }

  if (v0) {
    const size_t bi = (size_t)b * M_ + i0;
    max_ov[bi] = best0;  gt_assign[bi] = bk0;
    const bool  fg = (best0 >= FG_THRESH);
    const float d  = 1.0f - best0;
    const float fw = fg ? d * d : 0.0f;
    const float ow = (fw > 0.0f) ? 1.0f : 0.0f;
    ((float4*)inside_w)[bi]  = make_float4(fw, fw, fw, fw);
    ((float4*)outside_w)[bi] = make_float4(ow, ow, ow, ow);
  }
  if (v1) {
    const size_t bi = (size_t)b * M_ + i1;
    max_ov[bi] = best1;  gt_assign[bi] = bk1;
    const bool  fg = (best1 >= FG_THRESH);
    const float d  = 1.0f - best1;
    const float fw = fg ? d * d : 0.0f;
    const float ow = (fw > 0.0f) ? 1.0f : 0.0f;
    ((float4*)inside_w)[bi]  = make_float4(fw, fw, fw, fw);
    ((float4*)outside_w)[bi] = make_float4(ow, ow, ow, ow);
  }
}

// ---------------------------------------------------------------------------
// Kernel 2: one block per batch. Stable fg/bg partition via block-wide
// exclusive prefix sum of the fg flag, then deterministic scatter of
// rois_batch / labels_batch / bbox_targets to permuted positions.
// dest(i) = fg ? fgBefore(i) : fgCount + (i - fgBefore(i))   (bijection)
// ---------------------------------------------------------------------------
__global__ void __launch_bounds__(1024)
partition_scatter_kernel(const float* __restrict__ all_rois,
                         const float* __restrict__ gt_boxes,
                         const float* __restrict__ max_ov,
                         const int*   __restrict__ gt_assign,
                         float* __restrict__ rois_batch,    // (B,M,5)
                         float* __restrict__ labels_batch,  // (B,M)
                         float* __restrict__ bbox_targets)  // (B,M,4)
{
  __shared__ __align__(16) float sgt[K_ * 5];
  __shared__ int scan[1024];

  const int b   = blockIdx.x;
  const int tid = threadIdx.x;

  stage_gt_async(gt_boxes + (size_t)b * K_ * 5, sgt, tid);

  const int C     = (M_ + 1023) / 1024;   // 9 contiguous elems per thread
  const int start = tid * C;
  const int end   = min(start + C, M_);

  unsigned fgmask = 0u;                   // per-chunk fg flags cached
  int cnt = 0;
  for (int i = start, j = 0; i < end; ++i, ++j) {
    const bool fg = (max_ov[(size_t)b * M_ + i] >= FG_THRESH);
    fgmask |= (fg ? 1u : 0u) << j;
    cnt += fg ? 1 : 0;
  }

  scan[tid] = cnt;
  __syncthreads();
  // inclusive Hillis-Steele scan over 1024 per-thread counts
  for (int off = 1; off < 1024; off <<= 1) {
    const int add = (tid >= off) ? scan[tid - off] : 0;
    __syncthreads();
    scan[tid] += add;
    __syncthreads();
  }
  const int fgCount  = scan[1023];
  int       fgBefore = scan[tid] - cnt;   // exclusive prefix at chunk start

  for (int i = start, j = 0; i < end; ++i, ++j) {
    const size_t bi = (size_t)b * M_ + i;
    const bool   fg = ((fgmask >> j) & 1u) != 0u;
    const int    p  = fg ? fgBefore : (fgCount + (i - fgBefore));

    float rx1, ry1, rx2, ry2;
    if (i < N_) {
      const float* q = all_rois + ((size_t)b * N_ + i) * 5;
      rx1 = q[1]; ry1 = q[2]; rx2 = q[3]; ry2 = q[4];
    } else {
      const int k = i - N_;
      rx1 = sgt[k * 5 + 0]; ry1 = sgt[k * 5 + 1];
      rx2 = sgt[k * 5 + 2]; ry2 = sgt[k * 5 + 3];
    }

    const size_t bp = (size_t)b * M_ + p;
    float* r = rois_batch + bp * 5;
    r[0] = (float)b; r[1] = rx1; r[2] = ry1; r[3] = rx2; r[4] = ry2;

    const int   ga    = gt_assign[bi];
    const float label = fg ? sgt[ga * 5 + 4] : 0.0f;
    labels_batch[bp] = label;

    float4 t = make_float4(0.0f, 0.0f, 0.0f, 0.0f);
    if (label > 0.0f) {   // (labels_batch > 0) gate of the reference
      const float ew  = rx2 - rx1 + 1.0f, eh  = ry2 - ry1 + 1.0f;
      const float ecx = rx1 + 0.5f * ew,  ecy = ry1 + 0.5f * eh;
      const float gx1 = sgt[ga * 5 + 0], gy1 = sgt[ga * 5 + 1];
      const float gx2 = sgt[ga * 5 + 2], gy2 = sgt[ga * 5 + 3];
      const float gw  = gx2 - gx1 + 1.0f, gh  = gy2 - gy1 + 1.0f;
      const float gcx = gx1 + 0.5f * gw,  gcy = gy1 + 0.5f * gh;
      // (target - MEAN)/STD with MEAN=0, STD={0.1,0.1,0.2,0.2}
      t.x = ((gcx - ecx) / ew) / 0.1f;
      t.y = ((gcy - ecy) / eh) / 0.1f;
      t.z = logf(gw / ew) / 0.2f;
      t.w = logf(gh / eh) / 0.2f;
    }
    ((float4*)bbox_targets)[bp] = t;

    if (fg) fgBefore++;
  }
}

// ---------------------------------------------------------------------------
extern "C" void kernel_launch(void* const* d_in, const int* in_sizes, int n_in,
                              void* d_out, int out_size, void* d_ws, size_t ws_size,
                              hipStream_t stream) {
  const float* all_rois = (const float*)d_in[0];   // (B,N,5)
  const float* gt_boxes = (const float*)d_in[1];   // (B,K,5)

  float* out          = (float*)d_out;
  float* rois_batch   = out;                                   // B*M*5
  float* labels_batch = rois_batch + (size_t)B_ * M_ * 5;      // B*M
  float* bbox_targets = labels_batch + (size_t)B_ * M_;        // B*M*4
  float* inside_w     = bbox_targets + (size_t)B_ * M_ * 4;    // B*M*4
  float* outside_w    = inside_w + (size_t)B_ * M_ * 4;        // B*M*4

  float* max_ov    = (float*)d_ws;                             // B*M floats
  int*   gt_assign = (int*)(max_ov + (size_t)B_ * M_);         // B*M ints

  dim3 g1((M_ + ROIS_PER_BLOCK - 1) / ROIS_PER_BLOCK, B_);
  iou_argmax_kernel<<<g1, 256, 0, stream>>>(all_rois, gt_boxes, max_ov,
                                            gt_assign, inside_w, outside_w);
  partition_scatter_kernel<<<B_, 1024, 0, stream>>>(all_rois, gt_boxes, max_ov,
                                                    gt_assign, rois_batch,
                                                    labels_batch, bbox_targets);
}